// MultiHeadGAT_31782757990558
// MI455X (gfx1250) — compile-verified
//
#include <hip/hip_runtime.h>
#include <math.h>
#include <stdint.h>

// Problem constants (match reference)
#define BB    8
#define NN    1024
#define INF_  64
#define OUTF  32
#define HH    8
#define BH    (BB * HH)
#define ALPHA 0.2f
#define NEG_BIG (-9.0e15f)
#define LN_EPS  1e-5f
#define JCHUNK  256

typedef __attribute__((ext_vector_type(2))) float v2f;
typedef __attribute__((ext_vector_type(8))) float v8f;

static __device__ __forceinline__ v8f wmma_f32_4(v2f a, v2f b, v8f c) {
  // V_WMMA_F32_16X16X4_F32: D = A(16x4) x B(4x16) + C(16x16), fp32 exact
  return __builtin_amdgcn_wmma_f32_16x16x4_f32(
      /*neg_a=*/false, a, /*neg_b=*/false, b,
      /*c_mod=*/(short)0, c, /*reuse_a=*/false, /*reuse_b=*/false);
}

static __device__ __forceinline__ float lrelu(float x) {
  return x > 0.f ? x : ALPHA * x;
}

// CDNA5 async global->LDS copy (ASYNCcnt-tracked, no VGPR round trip).
// lds_byte_off: byte address within the workgroup's LDS allocation
// (low 32 bits of the flat shared-pointer, per ISA 10.2 aperture rules).
static __device__ __forceinline__ void async_copy_b128_to_lds(uint32_t lds_byte_off,
                                                              const void* gaddr) {
  asm volatile("global_load_async_to_lds_b128 %0, %1, off"
               :
               : "v"(lds_byte_off), "v"(gaddr)
               : "memory");
}
static __device__ __forceinline__ void wait_asynccnt0() {
  asm volatile("s_wait_asynccnt 0x0" ::: "memory");
}

// ---------------------------------------------------------------------------
// Kernel 1: Wh[bh][n][o] = h[b][n][:] @ W[hd][:][o]   (fp32 WMMA, K=64)
// One wave per 16x32 output tile.
// ---------------------------------------------------------------------------
__global__ __launch_bounds__(32) void k_proj(const float* __restrict__ h,
                                             const float* __restrict__ W,
                                             float* __restrict__ Wh) {
  const int bh  = blockIdx.x >> 6;   // 0..63
  const int it  = blockIdx.x & 63;   // i-tile (16 rows)
  const int b   = bh >> 3;
  const int hd  = bh & 7;
  const int lane = threadIdx.x;
  const int l15  = lane & 15;
  const int kh   = lane >> 4;        // K-half select (A/B 32-bit layout)
  const int row  = it * 16 + l15;

  const float* hrow = h + ((size_t)b * NN + row) * INF_;
  const float* Wm   = W + (size_t)hd * INF_ * OUTF;

  v8f acc0 = {};
  v8f acc1 = {};
  for (int ks = 0; ks < 16; ++ks) {
    const int k0 = ks * 4 + 2 * kh;
    v2f a;
    a.x = hrow[k0];
    a.y = hrow[k0 + 1];
    v2f b0, b1;
    b0.x = Wm[k0 * OUTF + l15];
    b0.y = Wm[(k0 + 1) * OUTF + l15];
    b1.x = Wm[k0 * OUTF + 16 + l15];
    b1.y = Wm[(k0 + 1) * OUTF + 16 + l15];
    acc0 = wmma_f32_4(a, b0, acc0);
    acc1 = wmma_f32_4(a, b1, acc1);
  }
  // C/D layout: VGPR v -> row v (lanes 0-15) or v+8 (lanes 16-31)
  float* out = Wh + (size_t)bh * NN * OUTF;
#pragma unroll
  for (int v = 0; v < 8; ++v) {
    const int rr = it * 16 + v + 8 * kh;
    out[rr * OUTF + l15]      = acc0[v];
    out[rr * OUTF + 16 + l15] = acc1[v];
  }
}

// ---------------------------------------------------------------------------
// Kernel 2: f1 = Wh . a1 , f2 = Wh . a2   (one thread per (bh,n) row)
// ---------------------------------------------------------------------------
__global__ __launch_bounds__(256) void k_fvec(const float* __restrict__ Wh,
                                              const float* __restrict__ a1,
                                              const float* __restrict__ a2,
                                              float* __restrict__ f1,
                                              float* __restrict__ f2) {
  const int idx = blockIdx.x * 256 + threadIdx.x;  // 0..BH*NN-1
  const int bh  = idx >> 10;
  const int hd  = bh & 7;
  const float* w  = Wh + (size_t)idx * OUTF;
  const float* v1 = a1 + hd * OUTF;
  const float* v2 = a2 + hd * OUTF;
  float s1 = 0.f, s2 = 0.f;
#pragma unroll
  for (int o = 0; o < OUTF; ++o) {
    const float wv = w[o];
    s1 += wv * v1[o];
    s2 += wv * v2[o];
  }
  f1[idx] = s1;
  f2[idx] = s2;
}

// ---------------------------------------------------------------------------
// Kernel 3: per-row softmax stats (max m, exp-sum s). One wave32 per row.
// Masked entries use the reference's finite -9e15 so a fully-masked row
// naturally yields uniform 1/N attention.
// ---------------------------------------------------------------------------
__global__ __launch_bounds__(256) void k_stats(const float* __restrict__ f1,
                                               const float* __restrict__ f2,
                                               const int* __restrict__ adj,
                                               float* __restrict__ mrow,
                                               float* __restrict__ srow) {
  const int wid  = blockIdx.x * 8 + (threadIdx.x >> 5);  // row id 0..BH*NN-1
  const int lane = threadIdx.x & 31;
  const int bh   = wid >> 10;
  const int i    = wid & 1023;
  const float f1v = f1[wid];
  const float* f2r = f2 + (size_t)bh * NN;
  const int* arow  = adj + (size_t)i * NN;

  float m = -INFINITY;
  for (int j = lane; j < NN; j += 32) {
    const float e = (arow[j] > 0) ? lrelu(f1v + f2r[j]) : NEG_BIG;
    m = fmaxf(m, e);
  }
#pragma unroll
  for (int off = 16; off > 0; off >>= 1) m = fmaxf(m, __shfl_xor(m, off, 32));

  float s = 0.f;
  for (int j = lane; j < NN; j += 32) {
    const float e = (arow[j] > 0) ? lrelu(f1v + f2r[j]) : NEG_BIG;
    s += __expf(e - m);
  }
#pragma unroll
  for (int off = 16; off > 0; off >>= 1) s += __shfl_xor(s, off, 32);

  if (lane == 0) {
    mrow[wid] = m;
    srow[wid] = s;
  }
}

// ---------------------------------------------------------------------------
// Kernel 4: h'[bh][i][o] = ELU( (1/s_i) * sum_j exp(e_ij - m_i) * Wh[j][o] )
// 8 waves/block, each wave owns 16 rows. Wh is staged into LDS in 32KB
// chunks with CDNA5 async global->LDS b128 copies (ASYNCcnt), then the
// attention probabilities are built on the fly into the WMMA A operand.
// 512 v_wmma_f32_16x16x4_f32 per wave.
// ---------------------------------------------------------------------------
__global__ __launch_bounds__(256) void k_attn(const float* __restrict__ Wh,
                                              const float* __restrict__ f1,
                                              const float* __restrict__ f2,
                                              const float* __restrict__ mrow,
                                              const float* __restrict__ srow,
                                              const int* __restrict__ adj,
                                              float* __restrict__ hp) {
  __shared__ __align__(16) float ldsWh[JCHUNK * OUTF];  // 32 KB
  __shared__ __align__(16) float ldsF2[JCHUNK];         // 1 KB

  const int bh   = blockIdx.x >> 3;
  const int iblk = blockIdx.x & 7;
  const int wv   = threadIdx.x >> 5;
  const int lane = threadIdx.x & 31;
  const int l15  = lane & 15;
  const int kh   = lane >> 4;
  const int rowbase = iblk * 128 + wv * 16;
  const int row     = rowbase + l15;   // A-operand row this lane feeds

  const float f1v = f1[bh * NN + row];
  const float mv  = mrow[bh * NN + row];
  const int* arow = adj + (size_t)row * NN;
  const float* whsrc = Wh + (size_t)bh * NN * OUTF;
  const float* f2src = f2 + (size_t)bh * NN;

  // Byte offsets of the LDS buffers inside the workgroup allocation
  // (low 32 bits of the flat address = LDS byte address).
  const uint32_t ldsWh_off = (uint32_t)(uintptr_t)(&ldsWh[0]);
  const uint32_t ldsF2_off = (uint32_t)(uintptr_t)(&ldsF2[0]);

  v8f acc0 = {};
  v8f acc1 = {};

  for (int jc = 0; jc < NN / JCHUNK; ++jc) {
    // --- async global -> LDS stage: 32KB Wh chunk + 1KB f2 chunk ---------
    const float* chunk = whsrc + (size_t)jc * JCHUNK * OUTF;
#pragma unroll
    for (int u = 0; u < (JCHUNK * OUTF) / (4 * 256); ++u) {   // 8 x b128/thread
      const int e16 = u * 256 + threadIdx.x;                  // 16B element id
      async_copy_b128_to_lds(ldsWh_off + e16 * 16, chunk + e16 * 4);
    }
    if (threadIdx.x < JCHUNK / 4) {                           // 64 x b128 total
      async_copy_b128_to_lds(ldsF2_off + threadIdx.x * 16,
                             f2src + jc * JCHUNK + threadIdx.x * 4);
    }
    wait_asynccnt0();     // this wave's async copies have landed in LDS
    __syncthreads();      // all waves' copies visible to everyone

    if (jc + 1 < NN / JCHUNK)  // warm L2 for next chunk
      __builtin_prefetch(whsrc + (size_t)(jc + 1) * JCHUNK * OUTF, 0, 1);

    for (int ks = 0; ks < JCHUNK / 4; ++ks) {
      const int jj = ks * 4 + 2 * kh;      // local K pair this lane supplies
      const int j  = jc * JCHUNK + jj;     // global column index
      v2f a;
      {
        const float e0 = (arow[j] > 0)     ? lrelu(f1v + ldsF2[jj])     : NEG_BIG;
        const float e1 = (arow[j + 1] > 0) ? lrelu(f1v + ldsF2[jj + 1]) : NEG_BIG;
        a.x = __expf(e0 - mv);
        a.y = __expf(e1 - mv);
      }
      v2f b0, b1;
      b0.x = ldsWh[jj * OUTF + l15];
      b0.y = ldsWh[(jj + 1) * OUTF + l15];
      b1.x = ldsWh[jj * OUTF + 16 + l15];
      b1.y = ldsWh[(jj + 1) * OUTF + 16 + l15];
      acc0 = wmma_f32_4(a, b0, acc0);
      acc1 = wmma_f32_4(a, b1, acc1);
    }
    __syncthreads();  // protect LDS before next chunk overwrite
  }

  // Normalize by row sum (note: C rows differ from this lane's A row),
  // apply ELU, store h'.
  float* out = hp + (size_t)bh * NN * OUTF;
#pragma unroll
  for (int v = 0; v < 8; ++v) {
    const int rr = rowbase + v + 8 * kh;
    const float sinv = 1.0f / srow[bh * NN + rr];
    const float x0 = acc0[v] * sinv;
    const float x1 = acc1[v] * sinv;
    out[rr * OUTF + l15]      = x0 > 0.f ? x0 : (__expf(x0) - 1.0f);
    out[rr * OUTF + 16 + l15] = x1 > 0.f ? x1 : (__expf(x1) - 1.0f);
  }
}

// ---------------------------------------------------------------------------
// Kernel 5: att_avg[b][i][j] = (1/H) sum_h exp(e_ij - m)/s  (recompute, no
// 256MB attention tensor, no atomics). One thread per output element.
// ---------------------------------------------------------------------------
__global__ __launch_bounds__(256) void k_attavg(const float* __restrict__ f1,
                                                const float* __restrict__ f2,
                                                const float* __restrict__ mrow,
                                                const float* __restrict__ srow,
                                                const int* __restrict__ adj,
                                                float* __restrict__ att_avg) {
  const size_t idx = (size_t)blockIdx.x * 256 + threadIdx.x;  // 0..B*N*N-1
  const int j = (int)(idx & 1023);
  const int i = (int)((idx >> 10) & 1023);
  const int b = (int)(idx >> 20);
  const int am = adj[(size_t)i * NN + j];
  float acc = 0.f;
#pragma unroll
  for (int hd = 0; hd < HH; ++hd) {
    const int r = (b * HH + hd) * NN + i;
    const float e = (am > 0) ? lrelu(f1[r] + f2[(b * HH + hd) * NN + j]) : NEG_BIG;
    acc += __expf(e - mrow[r]) / srow[r];
  }
  att_avg[idx] = acc * (1.0f / HH);
}

// ---------------------------------------------------------------------------
// Kernel 6: concat heads + LayerNorm. One 256-thread block per (b,n);
// thread t = hd*32+o matches the reference's transpose/reshape ordering.
// ---------------------------------------------------------------------------
__global__ __launch_bounds__(256) void k_layernorm(const float* __restrict__ hp,
                                                   const float* __restrict__ gamma,
                                                   const float* __restrict__ beta,
                                                   float* __restrict__ xout) {
  __shared__ float red[256];
  const int bn = blockIdx.x;   // b*N + n
  const int b  = bn >> 10;
  const int n  = bn & 1023;
  const int t  = threadIdx.x;
  const int hd = t >> 5;
  const int o  = t & 31;

  const float v = hp[((size_t)(b * HH + hd) * NN + n) * OUTF + o];

  red[t] = v;
  __syncthreads();
#pragma unroll
  for (int s = 128; s > 0; s >>= 1) {
    if (t < s) red[t] += red[t + s];
    __syncthreads();
  }
  const float mu = red[0] * (1.0f / 256.0f);
  __syncthreads();

  const float d = v - mu;
  red[t] = d * d;
  __syncthreads();
#pragma unroll
  for (int s = 128; s > 0; s >>= 1) {
    if (t < s) red[t] += red[t + s];
    __syncthreads();
  }
  const float var  = red[0] * (1.0f / 256.0f);
  const float rstd = rsqrtf(var + LN_EPS);

  xout[(size_t)bn * (HH * OUTF) + t] = d * rstd * gamma[t] + beta[t];
}

// ---------------------------------------------------------------------------
extern "C" void kernel_launch(void* const* d_in, const int* in_sizes, int n_in,
                              void* d_out, int out_size, void* d_ws, size_t ws_size,
                              hipStream_t stream) {
  (void)in_sizes; (void)n_in; (void)out_size; (void)ws_size;

  const float* h     = (const float*)d_in[0];  // [B,N,IN_F]
  const int*   adj   = (const int*)d_in[1];    // [N,N]
  const float* W     = (const float*)d_in[2];  // [H,IN_F,OUT_F]
  const float* a1    = (const float*)d_in[3];  // [H,OUT_F]
  const float* a2    = (const float*)d_in[4];  // [H,OUT_F]
  const float* gamma = (const float*)d_in[5];  // [H*OUT_F]
  const float* beta  = (const float*)d_in[6];  // [H*OUT_F]

  float* x       = (float*)d_out;                        // [B,N,H*OUT_F]
  float* att_avg = x + (size_t)BB * NN * (HH * OUTF);    // [B,N,N]

  // Workspace layout (floats): Wh | f1 | f2 | m | s | h'   (~17.8 MB)
  float* ws   = (float*)d_ws;
  size_t off  = 0;
  float* Wh   = ws + off; off += (size_t)BH * NN * OUTF;
  float* f1   = ws + off; off += (size_t)BH * NN;
  float* f2   = ws + off; off += (size_t)BH * NN;
  float* mrow = ws + off; off += (size_t)BH * NN;
  float* srow = ws + off; off += (size_t)BH * NN;
  float* hp   = ws + off;

  k_proj     <<<dim3(BH * (NN / 16)),        dim3(32),  0, stream>>>(h, W, Wh);
  k_fvec     <<<dim3(BH * NN / 256),         dim3(256), 0, stream>>>(Wh, a1, a2, f1, f2);
  k_stats    <<<dim3(BH * NN / 8),           dim3(256), 0, stream>>>(f1, f2, adj, mrow, srow);
  k_attn     <<<dim3(BH * 8),                dim3(256), 0, stream>>>(Wh, f1, f2, mrow, srow, adj, hp);
  k_attavg   <<<dim3(BB * NN * NN / 256),    dim3(256), 0, stream>>>(f1, f2, mrow, srow, adj, att_avg);
  k_layernorm<<<dim3(BB * NN),               dim3(256), 0, stream>>>(hp, gamma, beta, x);
}